// GroupGATLSTM_8693013807248
// MI455X (gfx1250) — compile-verified
//
#include <hip/hip_runtime.h>
#include <hip/hip_bf16.h>
#include <math.h>

// ---------------- problem constants ----------------
#define NN    500      // nodes
#define EE    16000    // edges (plus NN self loops appended)
#define BB    16
#define SS    64
#define TT    (BB*SS)  // 1024  (t = b*64 + s)
#define GATH  32
#define LH    128
#define G4    (4*LH)   // 512

typedef __attribute__((ext_vector_type(2))) float v2f;
typedef __attribute__((ext_vector_type(8))) float v8f;

// full-precision f32 WMMA: D(16x16) = A(16x4) * B(4x16) + C
__device__ inline v8f wmma_f32(v2f a, v2f b, v8f c) {
    return __builtin_amdgcn_wmma_f32_16x16x4_f32(false, a, false, b, (short)0, c, false, false);
}

// ---- CDNA5 async global->LDS copy (ASYNCcnt-tracked, 16B granule) ----
__device__ inline unsigned int lds_offset_of(const void* p) {
    // generic -> LDS(addrspace 3) pointer: 32-bit offset from LDS base
    return (unsigned int)(unsigned long long)
        (const __attribute__((address_space(3))) void*)p;
}
__device__ inline void async_copy_b128(void* lds_dst, const void* gsrc) {
    unsigned int loff = lds_offset_of(lds_dst);
    unsigned long long ga = (unsigned long long)gsrc;
    asm volatile("global_load_async_to_lds_b128 %0, %1, off"
                 :: "v"(loff), "v"(ga) : "memory");
}
__device__ inline void wait_async0() {
#if __has_builtin(__builtin_amdgcn_s_wait_asynccnt)
    __builtin_amdgcn_s_wait_asynccnt(0);
#else
    asm volatile("s_wait_asynccnt 0x0" ::: "memory");
#endif
}

// monotone float atomic-max (memory must be initialized to -inf)
__device__ inline void atomicMaxFloat(float* addr, float val) {
    if (val >= 0.0f) atomicMax((int*)addr, __float_as_int(val));
    else             atomicMin((unsigned int*)addr, (unsigned int)__float_as_int(val));
}

__device__ inline float sigmoidf_(float x) { return 1.0f / (1.0f + __expf(-x)); }

// ---------------- workspace layout (float offsets) ----------------
#define WS_XT    0u                      // [NN][TT]  x transposed: xt[n*TT + t]
#define WS_M     (WS_XT  + NN*TT)       // [NN][TT]  segment max
#define WS_SW    (WS_M   + NN*TT)       // [NN][TT]  segment sum of w
#define WS_SWX   (WS_SW  + NN*TT)       // [NN][TT]  segment sum of w*x_src
#define WS_AGG   (WS_SWX + NN*TT)       // [NN][TT]  agg
#define WS_PRE0  (WS_AGG + NN*TT)       // [TT][G4]  layer0 input projection
#define WS_H1    (WS_PRE0 + TT*G4)      // [TT][LH]  layer0 hidden sequence
#define WS_PRE1  (WS_H1  + TT*LH)       // [TT][G4]
#define WS_H2    (WS_PRE1 + TT*G4)      // [TT][LH]
#define WS_CSD   (WS_H2  + TT*LH)       // [2] c_src, c_dst

// ---------------- tiny setup: c_src, c_dst ----------------
__global__ void k_setup(const float* __restrict__ gw, const float* __restrict__ asrc,
                        const float* __restrict__ adst, float* __restrict__ csd) {
    if (threadIdx.x == 0) {
        float s = 0.f, d = 0.f;
        for (int k = 0; k < GATH; ++k) { s += gw[k] * asrc[k]; d += gw[k] * adst[k]; }
        csd[0] = s; csd[1] = d;
    }
}

// ---------------- transpose x -> xt[n][t], init reduction buffers ----------------
__global__ void k_init(const float* __restrict__ x, float* __restrict__ xt,
                       float* __restrict__ mmax, float* __restrict__ sw, float* __restrict__ swx) {
    int idx = blockIdx.x * 256 + threadIdx.x;
    if (idx >= NN * TT) return;
    int n = idx >> 10;          // TT == 1024
    int t = idx & (TT - 1);
    xt[idx]   = x[(size_t)t * NN + n];   // x flat: (b*S + s)*N + n = t*N + n
    mmax[idx] = __int_as_float(0xFF800000); // -inf
    sw[idx]   = 0.0f;
    swx[idx]  = 0.0f;
}

// ---------------- GAT pass 1: segment max over dst ----------------
__global__ void k_edge_max(const int* __restrict__ ei, const float* __restrict__ xt,
                           const float* __restrict__ csd, float* __restrict__ mmax) {
    int e = blockIdx.x;
    int src, dst;
    if (e < EE) { src = ei[e]; dst = ei[EE + e]; } else { src = dst = e - EE; }
    float cs = csd[0], cd = csd[1];
    const float* xs = xt + (size_t)src * TT;
    const float* xd = xt + (size_t)dst * TT;
    float* mrow = mmax + (size_t)dst * TT;
    for (int t = threadIdx.x; t < TT; t += 256) {
        float v = cs * xs[t] + cd * xd[t];
        float l = v > 0.0f ? v : 0.2f * v;  // leaky_relu 0.2
        atomicMaxFloat(&mrow[t], l);
    }
}

// ---------------- GAT pass 2: segment sums of w and w*x_src ----------------
__global__ void k_edge_sum(const int* __restrict__ ei, const float* __restrict__ xt,
                           const float* __restrict__ csd, const float* __restrict__ mmax,
                           float* __restrict__ sw, float* __restrict__ swx) {
    int e = blockIdx.x;
    int src, dst;
    if (e < EE) { src = ei[e]; dst = ei[EE + e]; } else { src = dst = e - EE; }
    float cs = csd[0], cd = csd[1];
    const float* xs = xt + (size_t)src * TT;
    const float* xd = xt + (size_t)dst * TT;
    const float* mrow = mmax + (size_t)dst * TT;
    float* swr  = sw  + (size_t)dst * TT;
    float* swxr = swx + (size_t)dst * TT;
    for (int t = threadIdx.x; t < TT; t += 256) {
        float xsv = xs[t];
        float v = cs * xsv + cd * xd[t];
        float l = v > 0.0f ? v : 0.2f * v;
        float w = __expf(l - mrow[t]);
        atomicAdd(&swr[t],  w);
        atomicAdd(&swxr[t], w * xsv);
    }
}

// ---------------- GAT finalize: agg = swx / sw ----------------
__global__ void k_agg(const float* __restrict__ swx, const float* __restrict__ sw,
                      float* __restrict__ agg) {
    int idx = blockIdx.x * 256 + threadIdx.x;
    if (idx >= NN * TT) return;
    agg[idx] = swx[idx] / sw[idx];
}

// ---------------- big fused GEMM: pre0[t][g] = sum_{n,k} relu(agg[n][t]*W[k]+bias[k]) * w_ih0[g][n*32+k]
// A (1024 x 16000) generated on the fly.  B-tiles stream via double-buffered
// GLOBAL_LOAD_ASYNC_TO_LDS_B128 (ASYNCcnt), overlapping HBM with f32 WMMA.
#define BK_PAD 36   // 32 + 4 pad: row stride 144B (16B aligned for B128, conflict-free frag reads)
__global__ __launch_bounds__(256) void k_gemm_pre0(
    const float* __restrict__ agg, const float* __restrict__ gw, const float* __restrict__ gb,
    const float* __restrict__ wih, const float* __restrict__ bih, const float* __restrict__ bhh,
    float* __restrict__ pre) {
    __shared__ __align__(16) float Bs[2][64][BK_PAD];  // [buf][N=64][K=32] row-major
    __shared__ float ags[2][64];                       // agg[n, t0..t0+63]
    __shared__ float Wk[GATH];
    __shared__ float Bk[GATH];

    const int tid  = threadIdx.x;
    const int lane = tid & 31, wid = tid >> 5;
    const int wm = wid & 3, wn = wid >> 2;     // 4 x 2 wave grid
    const int t0 = blockIdx.x * 64, g0 = blockIdx.y * 64;
    const int am = lane & 15;                  // matrix row / B col within tile
    const int ak = (lane >> 4) << 1;           // K sub-offset {0,2}
    const int cm = (lane >> 4) << 3;           // C row offset {0,8}

    if (tid < GATH) { Wk[tid] = gw[tid]; Bk[tid] = gb[tid]; }

    // per-thread async-copy assignment: row j (4 threads/row), 4-float chunks
    const int jj = tid >> 2;                   // 0..63
    const int k4 = (tid & 3) << 2;             // 0,4,8,12  (second chunk at +16)
    const float* grow = wih + (size_t)(g0 + jj) * (NN * GATH) + k4;

    // prologue: stream node 0 tile into buffer 0
    async_copy_b128(&Bs[0][jj][k4],      grow);
    async_copy_b128(&Bs[0][jj][k4 + 16], grow + 16);
    if (tid < 64) ags[0][tid] = agg[t0 + tid];
    wait_async0();
    __syncthreads();

    v8f acc0 = {}; v8f acc1 = {};
    for (int n = 0; n < NN; ++n) {
        const int cur = n & 1;
        if (n + 1 < NN) {
            // stream node n+1 tile into the other buffer while computing on this one
            const float* gnext = grow + (size_t)(n + 1) * GATH;
            async_copy_b128(&Bs[cur ^ 1][jj][k4],      gnext);
            async_copy_b128(&Bs[cur ^ 1][jj][k4 + 16], gnext + 16);
            if (tid < 64) ags[cur ^ 1][tid] = agg[(size_t)(n + 1) * TT + t0 + tid];
        }

        const float av = ags[cur][wm * 16 + am];
        const int nb0 = wn * 32 + am;
        #pragma unroll
        for (int kk = 0; kk < 8; ++kk) {
            const int k0 = kk * 4;
            v2f a, b0, b1;
            float p0 = av * Wk[k0 + ak]     + Bk[k0 + ak];
            float p1 = av * Wk[k0 + ak + 1] + Bk[k0 + ak + 1];
            a.x = p0 > 0.f ? p0 : 0.f;      // relu -> A fragment
            a.y = p1 > 0.f ? p1 : 0.f;
            b0.x = Bs[cur][nb0][k0 + ak];      b0.y = Bs[cur][nb0][k0 + ak + 1];
            b1.x = Bs[cur][nb0 + 16][k0 + ak]; b1.y = Bs[cur][nb0 + 16][k0 + ak + 1];
            acc0 = wmma_f32(a, b0, acc0);
            acc1 = wmma_f32(a, b1, acc1);
        }
        wait_async0();     // own wave's next-tile DMA done
        __syncthreads();   // all waves done with buffers; next-tile visible to all
    }
    // epilogue: + b_ih0 + b_hh0
    #pragma unroll
    for (int r = 0; r < 8; ++r) {
        const int t = t0 + wm * 16 + r + cm;
        const int g = g0 + wn * 32 + am;
        pre[(size_t)t * G4 + g]      = acc0[r] + bih[g]      + bhh[g];
        pre[(size_t)t * G4 + g + 16] = acc1[r] + bih[g + 16] + bhh[g + 16];
    }
}

// ---------------- generic C(MxN) = A(MxK) * B(NxK)^T (+bias1+bias2), f32 WMMA ----------------
__global__ __launch_bounds__(256) void k_gemm_abt(
    const float* __restrict__ A, int lda,
    const float* __restrict__ B, int ldb,
    const float* __restrict__ bias1, const float* __restrict__ bias2,
    float* __restrict__ C, int ldc, int M, int N, int K) {
    __shared__ float As[64][33];
    __shared__ float Bs[32][64];
    const int tid  = threadIdx.x;
    const int lane = tid & 31, wid = tid >> 5;
    const int wm = wid & 3, wn = wid >> 2;
    const int t0 = blockIdx.x * 64, g0 = blockIdx.y * 64;
    const int am = lane & 15;
    const int ak = (lane >> 4) << 1;
    const int cm = (lane >> 4) << 3;
    const int ii = tid >> 2;            // 0..63
    const int k4 = (tid & 3) << 3;      // 0,8,16,24

    v8f acc0 = {}; v8f acc1 = {};
    for (int kg = 0; kg < K; kg += 32) {
        __syncthreads();
        #pragma unroll
        for (int q = 0; q < 8; ++q) {
            int kk = k4 + q;
            int gi = t0 + ii, gk = kg + kk;
            As[ii][kk] = (gi < M && gk < K) ? A[(size_t)gi * lda + gk] : 0.f;
            int gj = g0 + ii;
            Bs[kk][ii] = (gj < N && gk < K) ? B[(size_t)gj * ldb + gk] : 0.f;
        }
        __syncthreads();
        #pragma unroll
        for (int kk = 0; kk < 8; ++kk) {
            const int k0 = kk * 4;
            v2f a, b0, b1;
            a.x = As[wm * 16 + am][k0 + ak];
            a.y = As[wm * 16 + am][k0 + ak + 1];
            const int nb = wn * 32 + am;
            b0.x = Bs[k0 + ak][nb];      b0.y = Bs[k0 + ak + 1][nb];
            b1.x = Bs[k0 + ak][nb + 16]; b1.y = Bs[k0 + ak + 1][nb + 16];
            acc0 = wmma_f32(a, b0, acc0);
            acc1 = wmma_f32(a, b1, acc1);
        }
    }
    #pragma unroll
    for (int r = 0; r < 8; ++r) {
        const int t = t0 + wm * 16 + r + cm;
        if (t >= M) continue;
        int g = g0 + wn * 32 + am;
        if (g < N) {
            float v = acc0[r];
            if (bias1) v += bias1[g];
            if (bias2) v += bias2[g];
            C[(size_t)t * ldc + g] = v;
        }
        g += 16;
        if (g < N) {
            float v = acc1[r];
            if (bias1) v += bias1[g];
            if (bias2) v += bias2[g];
            C[(size_t)t * ldc + g] = v;
        }
    }
}

// ---------------- serial LSTM scan: one workgroup, state in LDS, WMMA recurrent GEMM ----------------
__global__ __launch_bounds__(256) void k_lstm(
    const float* __restrict__ pre,   // [TT][G4]  (row t = b*64+s)
    const float* __restrict__ whh,   // [G4][LH]
    float* __restrict__ hout) {      // [TT][LH]
    __shared__ float Hs[BB][LH + 2]; // +2 pad: kills LDS bank conflicts on A-frag reads
    __shared__ float Cs[BB][LH];
    __shared__ float Gs[BB][G4];

    const int tid  = threadIdx.x;
    const int lane = tid & 31, wid = tid >> 5;  // 8 waves, wave wid owns g in [wid*64, wid*64+64)
    const int am = lane & 15;
    const int ak = (lane >> 4) << 1;
    const int cm = (lane >> 4) << 3;
    const int g0 = wid * 64;

    for (int idx = tid; idx < BB * LH; idx += 256) {
        int b = idx >> 7, h = idx & (LH - 1);
        Hs[b][h] = 0.f; Cs[b][h] = 0.f;
    }
    __syncthreads();

    for (int s = 0; s < SS; ++s) {
        // G = H(16x128) @ whh^T  -> wave computes 16 x 64 slice with 4 accumulators
        v8f acc[4] = {{}, {}, {}, {}};
        for (int kk = 0; kk < LH / 4; ++kk) {
            const int k0 = kk * 4;
            v2f a;
            a.x = Hs[am][k0 + ak];
            a.y = Hs[am][k0 + ak + 1];
            #pragma unroll
            for (int tt = 0; tt < 4; ++tt) {
                const int g = g0 + tt * 16 + am;
                v2f b;
                b.x = whh[(size_t)g * LH + k0 + ak];
                b.y = whh[(size_t)g * LH + k0 + ak + 1];
                acc[tt] = wmma_f32(a, b, acc[tt]);
            }
        }
        // spill G (+ pre) to LDS
        #pragma unroll
        for (int tt = 0; tt < 4; ++tt) {
            const int g = g0 + tt * 16 + am;
            #pragma unroll
            for (int r = 0; r < 8; ++r) {
                const int m = r + cm;                 // batch row
                Gs[m][g] = acc[tt][r] + pre[(size_t)(m * SS + s) * G4 + g];
            }
        }
        __syncthreads();
        // gates (elementwise) + state update + stream h out
        for (int idx = tid; idx < BB * LH; idx += 256) {
            int b = idx >> 7, h = idx & (LH - 1);
            float gi = Gs[b][h];
            float gf = Gs[b][LH + h];
            float gg = Gs[b][2 * LH + h];
            float go = Gs[b][3 * LH + h];
            float c  = sigmoidf_(gf) * Cs[b][h] + sigmoidf_(gi) * tanhf(gg);
            float nh = sigmoidf_(go) * tanhf(c);
            Cs[b][h] = c;
            Hs[b][h] = nh;
            hout[(size_t)(b * SS + s) * LH + h] = nh;
        }
        __syncthreads();
    }
}

// ---------------- host launcher ----------------
extern "C" void kernel_launch(void* const* d_in, const int* in_sizes, int n_in,
                              void* d_out, int out_size, void* d_ws, size_t ws_size,
                              hipStream_t stream) {
    (void)in_sizes; (void)n_in; (void)out_size; (void)ws_size;
    const float* x      = (const float*)d_in[0];
    const int*   ei     = (const int*)  d_in[1];
    const float* gat_W  = (const float*)d_in[2];
    const float* a_src  = (const float*)d_in[3];
    const float* a_dst  = (const float*)d_in[4];
    const float* gat_b  = (const float*)d_in[5];
    const float* w_ih0  = (const float*)d_in[6];
    const float* w_hh0  = (const float*)d_in[7];
    const float* b_ih0  = (const float*)d_in[8];
    const float* b_hh0  = (const float*)d_in[9];
    const float* w_ih1  = (const float*)d_in[10];
    const float* w_hh1  = (const float*)d_in[11];
    const float* b_ih1  = (const float*)d_in[12];
    const float* b_hh1  = (const float*)d_in[13];
    const float* head_W = (const float*)d_in[14];
    const float* head_b = (const float*)d_in[15];
    float* out = (float*)d_out;
    float* ws  = (float*)d_ws;

    float* xt   = ws + WS_XT;
    float* mm   = ws + WS_M;
    float* sw   = ws + WS_SW;
    float* swx  = ws + WS_SWX;
    float* agg  = ws + WS_AGG;
    float* pre0 = ws + WS_PRE0;
    float* h1   = ws + WS_H1;
    float* pre1 = ws + WS_PRE1;
    float* h2   = ws + WS_H2;
    float* csd  = ws + WS_CSD;

    const int nt_blocks = (NN * TT + 255) / 256;   // 2000

    k_setup<<<1, 32, 0, stream>>>(gat_W, a_src, a_dst, csd);
    k_init<<<nt_blocks, 256, 0, stream>>>(x, xt, mm, sw, swx);
    k_edge_max<<<EE + NN, 256, 0, stream>>>(ei, xt, csd, mm);
    k_edge_sum<<<EE + NN, 256, 0, stream>>>(ei, xt, csd, mm, sw, swx);
    k_agg<<<nt_blocks, 256, 0, stream>>>(swx, sw, agg);

    // layer 0: fused GAT-feature x w_ih0^T GEMM (K = 16000 generated on the fly,
    // w_ih0 streamed via async-to-LDS double buffering)
    k_gemm_pre0<<<dim3(TT / 64, G4 / 64), 256, 0, stream>>>(agg, gat_W, gat_b,
                                                            w_ih0, b_ih0, b_hh0, pre0);
    k_lstm<<<1, 256, 0, stream>>>(pre0, w_hh0, h1);

    // layer 1 input projection: (1024x128) x (512x128)^T
    k_gemm_abt<<<dim3(TT / 64, G4 / 64), 256, 0, stream>>>(h1, LH, w_ih1, LH,
                                                           b_ih1, b_hh1, pre1, G4,
                                                           TT, G4, LH);
    k_lstm<<<1, 256, 0, stream>>>(pre1, w_hh1, h2);

    // head: z = h2[:, 63, :]  ->  out(16x500) = z x head_W^T + head_b
    k_gemm_abt<<<dim3(1, (NN + 63) / 64), 256, 0, stream>>>(h2 + 63 * LH, SS * LH,
                                                            head_W, LH,
                                                            head_b, nullptr, out, NN,
                                                            BB, NN, LH);
}